// Aggregator_62715112456964
// MI455X (gfx1250) — compile-verified
//
#include <hip/hip_runtime.h>

typedef float v2f __attribute__((ext_vector_type(2)));
typedef float v8f __attribute__((ext_vector_type(8)));

#define N_NODES   50000
#define DIM       256
#define NEG_SLOPE 0.01f
#define LDS_PITCH 260   // 260 % 64 == 4 -> conflict-free fragment reads
#define EDGES_PER_BLOCK 4

// ---------------- kernel 1: zero the side/output buffer ----------------
__global__ __launch_bounds__(256) void zero_f4(float* __restrict__ out, int n4) {
  int i = blockIdx.x * blockDim.x + threadIdx.x;
  if (i < n4) {
    float4 z = make_float4(0.f, 0.f, 0.f, 0.f);
    reinterpret_cast<float4*>(out)[i] = z;
  }
}

// ---------------- kernel 2: SpMM scatter (side += val * x[col]) ----------------
__global__ __launch_bounds__(256) void spmm_scatter(
    const int* __restrict__ erow, const int* __restrict__ ecol,
    const float* __restrict__ eval, const float* __restrict__ x,
    float* side, int n_edges) {
  const int t  = threadIdx.x;            // dim 0..255
  const int e0 = blockIdx.x * EDGES_PER_BLOCK;
#pragma unroll
  for (int i = 0; i < EDGES_PER_BLOCK; ++i) {
    int e = e0 + i;
    if (e < n_edges) {
      int   r = erow[e];                 // uniform per block -> scalar loads
      int   c = ecol[e];
      float v = eval[e];
      float m = v * x[(size_t)c * DIM + t];
      atomicAdd(&side[(size_t)r * DIM + t], m);
    }
  }
}

// ---------------- kernel 3: fused dual GEMM + bias + LeakyReLU + add ----------------
// out (== side on entry): block reads its own 16-row strip, then overwrites it.
__global__ __launch_bounds__(256) void fused_dual_gemm(
    const float* __restrict__ ego,
    const float* __restrict__ W1, const float* __restrict__ b1,
    const float* __restrict__ W2, const float* __restrict__ b2,
    float* out) {
  __shared__ float a1[16 * LDS_PITCH];   // ego + side
  __shared__ float a2[16 * LDS_PITCH];   // ego * side

  const int rowbase = blockIdx.x * 16;
  const int t = threadIdx.x;

  // Stage the 16x256 strip: 1024 float4 loads across 256 threads (4 each)
#pragma unroll
  for (int j = 0; j < 4; ++j) {
    int i4 = t + 256 * j;                // 0..1023
    int r  = i4 >> 6;                    // row in strip
    int c4 = i4 & 63;                    // float4 column
    float4 e = reinterpret_cast<const float4*>(ego + (size_t)(rowbase + r) * DIM)[c4];
    float4 s = reinterpret_cast<const float4*>(out + (size_t)(rowbase + r) * DIM)[c4];
    float* p1 = &a1[r * LDS_PITCH + c4 * 4];
    float* p2 = &a2[r * LDS_PITCH + c4 * 4];
    p1[0] = e.x + s.x; p1[1] = e.y + s.y; p1[2] = e.z + s.z; p1[3] = e.w + s.w;
    p2[0] = e.x * s.x; p2[1] = e.y * s.y; p2[2] = e.z * s.z; p2[3] = e.w * s.w;
  }
  __syncthreads();

  const int lane = t & 31;
  const int wave = t >> 5;               // 0..7
  const int m    = lane & 15;            // A row / B col / D col
  const int half = lane >> 4;            // K-half split (ISA 7.12.2)

  const float* a1p = &a1[m * LDS_PITCH + half * 2];
  const float* a2p = &a2[m * LDS_PITCH + half * 2];

#pragma unroll
  for (int cti = 0; cti < 2; ++cti) {
    const int colbase = (wave + cti * 8) * 16;
    const int jg = colbase + m;          // row of W (since out = x @ W^T)
    const float* w1p = W1 + (size_t)jg * DIM + half * 2;
    const float* w2p = W2 + (size_t)jg * DIM + half * 2;

    v8f acc1 = {};
    v8f acc2 = {};
#pragma unroll 8
    for (int k = 0; k < DIM; k += 4) {
      v2f fa1 = *reinterpret_cast<const v2f*>(a1p + k);
      v2f fa2 = *reinterpret_cast<const v2f*>(a2p + k);
      v2f fw1 = *reinterpret_cast<const v2f*>(w1p + k);
      v2f fw2 = *reinterpret_cast<const v2f*>(w2p + k);
      acc1 = __builtin_amdgcn_wmma_f32_16x16x4_f32(
          false, fa1, false, fw1, (short)0, acc1, false, false);
      acc2 = __builtin_amdgcn_wmma_f32_16x16x4_f32(
          false, fa2, false, fw2, (short)0, acc2, false, false);
    }

    const float bias1 = b1[jg];
    const float bias2 = b2[jg];
    float* op = out + (size_t)(rowbase + half * 8) * DIM + colbase + m;
#pragma unroll
    for (int v = 0; v < 8; ++v) {
      float s1 = acc1[v] + bias1;
      s1 = (s1 >= 0.f) ? s1 : NEG_SLOPE * s1;
      float s2 = acc2[v] + bias2;
      s2 = (s2 >= 0.f) ? s2 : NEG_SLOPE * s2;
      op[(size_t)v * DIM] = s1 + s2;
    }
  }
}

extern "C" void kernel_launch(void* const* d_in, const int* in_sizes, int n_in,
                              void* d_out, int out_size, void* d_ws, size_t ws_size,
                              hipStream_t stream) {
  const float* ego  = (const float*)d_in[0];
  const int*   erow = (const int*)  d_in[1];
  const int*   ecol = (const int*)  d_in[2];
  const float* eval = (const float*)d_in[3];
  const float* W1   = (const float*)d_in[4];
  const float* b1   = (const float*)d_in[5];
  const float* W2   = (const float*)d_in[6];
  const float* b2   = (const float*)d_in[7];
  float* out = (float*)d_out;
  const int n_edges = in_sizes[1];

  const int n4 = (N_NODES * DIM) / 4;
  zero_f4<<<(n4 + 255) / 256, 256, 0, stream>>>(out, n4);

  const int nblk = (n_edges + EDGES_PER_BLOCK - 1) / EDGES_PER_BLOCK;
  spmm_scatter<<<nblk, 256, 0, stream>>>(erow, ecol, eval, ego, out, n_edges);

  fused_dual_gemm<<<N_NODES / 16, 256, 0, stream>>>(ego, W1, b1, W2, b2, out);
}